// GraphSAGE_34196529610766
// MI455X (gfx1250) — compile-verified
//
#include <hip/hip_runtime.h>

typedef __attribute__((ext_vector_type(2))) float v2f;
typedef __attribute__((ext_vector_type(8))) float v8f;

// ---------------------------------------------------------------------------
// Zero a float buffer (workspace is poisoned; must re-zero every call).
// ---------------------------------------------------------------------------
__global__ void zero_f32_kernel(float* __restrict__ p, long long n) {
    long long i = (long long)blockIdx.x * blockDim.x + threadIdx.x;
    if (i < n) p[i] = 0.0f;
}

// ---------------------------------------------------------------------------
// Degree: cnt[dst[e]] += 1  (float; exact up to 2^24, E = 1.2M so fine).
// Computed once, reused by both layers (depends only on dst).
// ---------------------------------------------------------------------------
__global__ void degree_kernel(const long long* __restrict__ dst,
                              float* __restrict__ cnt, int E) {
    int e = blockIdx.x * blockDim.x + threadIdx.x;
    if (e < E) atomicAdd(&cnt[dst[e]], 1.0f);
}

// ---------------------------------------------------------------------------
// Edge scatter-add: agg[dst][d] += feat[src][d].
// One thread per (edge, dim); a wave covers 32 contiguous dims of one edge:
// coalesced 128B gather + same-cacheline atomic burst. All traffic L2-resident
// (feat: 25.6 MB, agg: 25.6 MB << 192 MB L2).
// ---------------------------------------------------------------------------
__global__ void scatter_kernel(const float* __restrict__ feat,
                               const long long* __restrict__ src,
                               const long long* __restrict__ dst,
                               float* __restrict__ agg, long long total) {
    long long tid = (long long)blockIdx.x * blockDim.x + threadIdx.x;
    if (tid >= total) return;
    int e = (int)(tid >> 6);
    int d = (int)(tid & 63);
    long long s = src[e];
    long long t = dst[e];
    atomicAdd(&agg[t * 64 + d], feat[s * 64 + d]);
}

// ---------------------------------------------------------------------------
// Fused SAGEConv: out = relu?( (agg/deg) @ Wl + b + xin @ Wr )
// One wave per 16-row tile, 4 C-tiles (16x16 f32 each), K=64 in 16 steps of 4.
// Uses V_WMMA_F32_16X16X4_F32 (full fp32 precision, matches reference).
//
// Fragment layouts (wave32):
//   A 16x4 f32 : lane L, VGPR j -> M = L%16, K = 2*(L/16) + j
//   B 4x16 f32 : lane L, VGPR j -> K = 2*(L/16) + j, N = L%16
//   C 16x16 f32: lane L, VGPR i -> M = i + 8*(L/16), N = L%16
// ---------------------------------------------------------------------------
__global__ void sage_wmma_kernel(const float* __restrict__ agg,   // N x 64 (sums)
                                 const float* __restrict__ cnt,   // N
                                 const float* __restrict__ xin,   // N x 64
                                 const float* __restrict__ Wl,    // 64 x 64
                                 const float* __restrict__ bias,  // 64
                                 const float* __restrict__ Wr,    // 64 x 64
                                 float* __restrict__ out,         // N x 64
                                 int N, int relu) {
    const int lane  = threadIdx.x & 31;
    const int wave  = threadIdx.x >> 5;
    const int tile  = blockIdx.x * (blockDim.x >> 5) + wave;
    const int row0  = tile * 16;
    if (row0 >= N) return;  // wave-uniform: EXEC stays all-1s for WMMA

    const int mrow  = lane & 15;   // A row / B,C column within tile
    const int khalf = lane >> 4;   // 0 or 1

    int row = row0 + mrow;
    if (row >= N) row = N - 1;     // clamp (N % 16 == 0 here, safety only)

    const float deg = cnt[row];
    const float inv = 1.0f / fmaxf(deg, 1.0f);

    // Accumulators initialized with bias: every VGPR of lane L is column
    // mrow + 16*t, so broadcast b[col] into all 8 components.
    v8f acc[4];
#pragma unroll
    for (int t = 0; t < 4; ++t) {
        float bv = bias[t * 16 + mrow];
#pragma unroll
        for (int i = 0; i < 8; ++i) acc[t][i] = bv;
    }

    const float* aggRow = agg + (size_t)row * 64;
    const float* xRow   = xin + (size_t)row * 64;

#pragma unroll 4
    for (int kk = 0; kk < 16; ++kk) {
        const int kbase = kk * 4 + khalf * 2;

        v2f aAgg, aX;
        aAgg[0] = aggRow[kbase]     * inv;   // fold mean-divide into A
        aAgg[1] = aggRow[kbase + 1] * inv;
        aX[0]   = xRow[kbase];
        aX[1]   = xRow[kbase + 1];

        const float* wl = Wl + (size_t)kbase * 64;
        const float* wr = Wr + (size_t)kbase * 64;

#pragma unroll
        for (int t = 0; t < 4; ++t) {
            const int col = t * 16 + mrow;
            v2f bl, br;
            bl[0] = wl[col];      bl[1] = wl[64 + col];
            br[0] = wr[col];      br[1] = wr[64 + col];
            acc[t] = __builtin_amdgcn_wmma_f32_16x16x4_f32(
                false, aAgg, false, bl, (short)0, acc[t], false, false);
            acc[t] = __builtin_amdgcn_wmma_f32_16x16x4_f32(
                false, aX, false, br, (short)0, acc[t], false, false);
        }
    }

    // Store: lane L, VGPR i -> row (row0 + i + 8*khalf), col (t*16 + mrow)
#pragma unroll
    for (int t = 0; t < 4; ++t) {
#pragma unroll
        for (int i = 0; i < 8; ++i) {
            float v = acc[t][i];
            if (relu) v = fmaxf(v, 0.0f);
            int r = row0 + khalf * 8 + i;
            if (r < N) out[(size_t)r * 64 + t * 16 + mrow] = v;
        }
    }
}

// ---------------------------------------------------------------------------
// Host-side launch. Inputs (dict order):
//   0: x (N*64 f32)   1: edge_index (2*E i64)
//   2: W1_l  3: b1  4: W1_r   5: W2_l  6: b2  7: W2_r
// Workspace: agg (N*64 f32) | cnt (N f32) | h (N*64 f32)  ~= 51.6 MB
// ---------------------------------------------------------------------------
extern "C" void kernel_launch(void* const* d_in, const int* in_sizes, int n_in,
                              void* d_out, int out_size, void* d_ws, size_t ws_size,
                              hipStream_t stream) {
    const float*     x    = (const float*)d_in[0];
    const long long* ei   = (const long long*)d_in[1];
    const float*     W1l  = (const float*)d_in[2];
    const float*     b1   = (const float*)d_in[3];
    const float*     W1r  = (const float*)d_in[4];
    const float*     W2l  = (const float*)d_in[5];
    const float*     b2   = (const float*)d_in[6];
    const float*     W2r  = (const float*)d_in[7];
    float*           out  = (float*)d_out;

    const int N = in_sizes[0] / 64;
    const int E = in_sizes[1] / 2;
    const long long* srcIdx = ei;        // edge_index[0]
    const long long* dstIdx = ei + E;    // edge_index[1]

    float* agg = (float*)d_ws;
    float* cnt = agg + (size_t)N * 64;
    float* h   = cnt + N;

    const long long aggN  = (long long)N * 64;
    const long long edgeW = (long long)E * 64;
    const int tiles = (N + 15) / 16;
    const int wavesPerBlock = 4;
    const dim3 gemmGrid((tiles + wavesPerBlock - 1) / wavesPerBlock);
    const dim3 gemmBlock(32 * wavesPerBlock);

    // ---- Layer 1 ----
    zero_f32_kernel<<<(unsigned)((aggN + N + 511) / 512), 512, 0, stream>>>(agg, aggN + N); // agg+cnt contiguous
    degree_kernel<<<(E + 255) / 256, 256, 0, stream>>>(dstIdx, cnt, E);
    scatter_kernel<<<(unsigned)((edgeW + 255) / 256), 256, 0, stream>>>(x, srcIdx, dstIdx, agg, edgeW);
    sage_wmma_kernel<<<gemmGrid, gemmBlock, 0, stream>>>(agg, cnt, x, W1l, b1, W1r, h, N, 1);

    // ---- Layer 2 (reuse cnt; re-zero agg only) ----
    zero_f32_kernel<<<(unsigned)((aggN + 511) / 512), 512, 0, stream>>>(agg, aggN);
    scatter_kernel<<<(unsigned)((edgeW + 255) / 256), 256, 0, stream>>>(h, srcIdx, dstIdx, agg, edgeW);
    sage_wmma_kernel<<<gemmGrid, gemmBlock, 0, stream>>>(agg, cnt, h, W2l, b2, W2r, out, N, 0);
}